// MS_AR_48816598286939
// MI455X (gfx1250) — compile-verified
//
#include <hip/hip_runtime.h>

// ---------------------------------------------------------------------------
// Reference collapses to a 32-vector linear recurrence:
//   s_t   = obs2[t] - bate*obs1[t]
//   c_j(t)= exp(-s_t * inv2v[j]),  inv2v = 1/(2 var)
//   R[i,j]= exp((mean[j]-bate*mean[i])*inv2v[j]) / sqrt(2*pi*var[j])
//   M  = aij .* R      (Mt = M^T stored row-major as Mt[j][i])
//   M2 = aij .* R^2
//   h_t = D_{c_t} M^T h_{t-1},  h_{-1} = pi
//   loss = sum_t  (h_{t-1}^T (M2 c_t^2)) / (1^T h_t)
//
// Blocked scan (B=16 steps/block):
//   P_k = A_{t0+k} ... A_{t0},  A = D_c M^T  -> 32x32 WMMA prefix products
//   qhat_k = Phat_{k-1}^T u_k (u = M2 c^2), rhat_k = colsum(Phat_k),
//   sigma_k = per-step renormalizer;  loss_{t0+k} = (qhat.h)/(rhat.h)/sigma_k
//   Pass 2 (single wave): sequential over blocks, h advance via Pend.
// ---------------------------------------------------------------------------

typedef float v2f __attribute__((ext_vector_type(2)));
typedef float v8f __attribute__((ext_vector_type(8)));

#define BSTEP 16
#define PI_F 3.1415926f

// ---------------- kernel 0: precompute Mt (= M^T) and M2 --------------------
__global__ void ms_ar_precompute(const float* __restrict__ mean,
                                 const float* __restrict__ var,
                                 const float* __restrict__ batep,
                                 const float* __restrict__ aij,
                                 float* __restrict__ Mtg,
                                 float* __restrict__ M2g) {
    int tid = threadIdx.x;          // 1024 threads
    int i = tid >> 5;               // source state
    int j = tid & 31;               // dest state
    float bate  = batep[0];
    float vj    = var[j];
    float inv2v = 0.5f / vj;
    float nrm   = rsqrtf(2.0f * PI_F * vj);
    float R  = __expf((mean[j] - bate * mean[i]) * inv2v) * nrm;
    float a  = aij[i * 32 + j];
    Mtg[j * 32 + i] = R * a;        // Mt[j][i] = M[i][j]
    M2g[i * 32 + j] = R * R * a;    // M2[i][j]
}

// ---------------- kernel 1: per-block WMMA prefix products ------------------
__global__ __launch_bounds__(128)
void ms_ar_pass1(const float* __restrict__ obs2,
                 const float* __restrict__ obs1,
                 const float* __restrict__ var,
                 const float* __restrict__ batep,
                 const float* __restrict__ Mtg,
                 const float* __restrict__ M2g,
                 float* __restrict__ qhatg,
                 float* __restrict__ rhatg,
                 float* __restrict__ sigg,
                 float* __restrict__ pendg,
                 int T) {
    __shared__ float Mt[1024];
    __shared__ float M2s[1024];
    __shared__ float P[1024];      // current prefix product Phat
    __shared__ float rawb[1024];   // unnormalized next product
    __shared__ float cl[32];       // c_j(t)
    __shared__ float c2l[32];      // c_j(t)^2
    __shared__ float uu[32];       // u = M2 * c^2
    __shared__ float il[32];       // inv2v
    __shared__ float sarr[BSTEP];  // s_t for this block
    __shared__ float red[128];
    __shared__ float sigsh;

    const int tid  = threadIdx.x;
    const int b    = blockIdx.x;
    const int t0   = b * BSTEP;
    const int w    = tid >> 5;
    const int lane = tid & 31;
    const int half = lane >> 4;     // 0: lanes 0-15, 1: lanes 16-31
    const int r    = lane & 15;
    const int tr   = w >> 1;        // C row tile
    const int tc   = w & 1;         // C col tile
    const float bate = batep[0];

    for (int e = tid; e < 1024; e += 128) {
        Mt[e]  = Mtg[e];
        M2s[e] = M2g[e];
        P[e]   = ((e >> 5) == (e & 31)) ? 1.0f : 0.0f;   // identity
    }
    if (tid < 32) il[tid] = 0.5f / var[tid];
    if (tid < BSTEP) {
        int t = t0 + tid;
        sarr[tid] = (t < T) ? (obs2[t] - bate * obs1[t]) : 0.0f;
    }
    __syncthreads();

    int kmax = T - t0; if (kmax > BSTEP) kmax = BSTEP;

    for (int k = 0; k < kmax; ++k) {
        // --- c, c^2 for this step ---
        if (tid < 32) {
            float cv = __expf(-sarr[k] * il[tid]);
            cl[tid]  = cv;
            c2l[tid] = cv * cv;
        }
        __syncthreads();
        // --- u[i] = sum_j M2[i][j] c2[j]  (split over 4 waves) ---
        {
            int i = tid & 31, jb = (tid >> 5) * 8;
            float p = 0.0f;
            #pragma unroll
            for (int j = jb; j < jb + 8; ++j) p += M2s[i * 32 + j] * c2l[j];
            red[tid] = p;
        }
        __syncthreads();
        if (tid < 32) uu[tid] = red[tid] + red[tid + 32] + red[tid + 64] + red[tid + 96];
        __syncthreads();
        // --- qhat[i] = sum_j Phat_{k-1}[j][i] u[j]  (k=0: P=I -> q=u) ---
        {
            int i = tid & 31, jb = (tid >> 5) * 8;
            float p = 0.0f;
            #pragma unroll
            for (int j = jb; j < jb + 8; ++j) p += P[j * 32 + i] * uu[j];
            red[tid] = p;
        }
        __syncthreads();
        if (tid < 32)
            qhatg[((size_t)b * BSTEP + k) * 32 + tid] =
                red[tid] + red[tid + 32] + red[tid + 64] + red[tid + 96];
        __syncthreads();

        // --- WMMA: Y = Mt x Phat_{k-1}  (32x32 = 2x2 tiles, K in 8 chunks) ---
        v8f acc = {0.f, 0.f, 0.f, 0.f, 0.f, 0.f, 0.f, 0.f};
        #pragma unroll
        for (int kk = 0; kk < 8; ++kk) {
            const int k0 = 4 * kk;
            v2f a, bb;
            // A tile 16x4: lane<16 -> K=k0,k0+1 ; lane>=16 -> K=k0+2,k0+3
            a.x = Mt[(16 * tr + r) * 32 + k0 + 2 * half + 0];
            a.y = Mt[(16 * tr + r) * 32 + k0 + 2 * half + 1];
            // B tile 4x16: same K split across lane halves / VGPRs
            bb.x = P[(k0 + 2 * half + 0) * 32 + 16 * tc + r];
            bb.y = P[(k0 + 2 * half + 1) * 32 + 16 * tc + r];
            acc = __builtin_amdgcn_wmma_f32_16x16x4_f32(
                false, a, false, bb, (short)0, acc, false, false);
        }
        // raw[j][i] = c[j] * Y[j][i]
        #pragma unroll
        for (int vr = 0; vr < 8; ++vr) {
            int row = 16 * tr + vr + 8 * half;   // global j
            int col = 16 * tc + r;               // global i
            rawb[row * 32 + col] = acc[vr] * cl[row];
        }
        __syncthreads();

        // --- sigma = sum(raw); Phat_k = raw / sigma ---
        {
            float p = 0.0f;
            for (int e = tid; e < 1024; e += 128) p += rawb[e];
            red[tid] = p;
        }
        __syncthreads();
        if (tid < 32) {
            float s2 = red[tid] + red[tid + 32] + red[tid + 64] + red[tid + 96];
            #pragma unroll
            for (int m = 16; m >= 1; m >>= 1) s2 += __shfl_xor(s2, m);
            if (tid == 0) sigsh = s2;
        }
        __syncthreads();
        {
            float inv = 1.0f / sigsh;
            for (int e = tid; e < 1024; e += 128) P[e] = rawb[e] * inv;
            if (tid == 0) sigg[(size_t)b * BSTEP + k] = sigsh;
        }
        __syncthreads();
        // --- rhat[i] = colsum(Phat_k) ---
        {
            int i = tid & 31, jb = (tid >> 5) * 8;
            float p = 0.0f;
            #pragma unroll
            for (int j = jb; j < jb + 8; ++j) p += P[j * 32 + i];
            red[tid] = p;
        }
        __syncthreads();
        if (tid < 32)
            rhatg[((size_t)b * BSTEP + k) * 32 + tid] =
                red[tid] + red[tid + 32] + red[tid + 64] + red[tid + 96];
        __syncthreads();
    }
    // block-end product for the sequential pass
    for (int e = tid; e < 1024; e += 128) pendg[(size_t)b * 1024 + e] = P[e];
}

// ---------------- kernel 2: sequential cross-block pass (1 wave) ------------
__global__ __launch_bounds__(32)
void ms_ar_pass2(const float* __restrict__ pi,
                 const float* __restrict__ qhatg,
                 const float* __restrict__ rhatg,
                 const float* __restrict__ sigg,
                 const float* __restrict__ pendg,
                 float* __restrict__ out,
                 int NB, int T) {
    __shared__ float hl[32];
    const int lane = threadIdx.x;
    hl[lane] = pi[lane];
    __syncthreads();

    float lacc = 0.0f;
    for (int b = 0; b < NB; ++b) {
        int kmax = T - b * BSTEP; if (kmax > BSTEP) kmax = BSTEP;
        // lanes 0-15: numer_k = qhat_k . h ; lanes 16-31: denom_k = rhat_k . h
        const int k = lane & 15;
        const float* vec = (lane < 16)
            ? &qhatg[((size_t)b * BSTEP + k) * 32]
            : &rhatg[((size_t)b * BSTEP + k) * 32];
        float acc = 0.0f;
        #pragma unroll 4
        for (int i = 0; i < 32; ++i) acc += vec[i] * hl[i];
        float other = __shfl_xor(acc, 16);
        if (lane < 16 && k < kmax)
            lacc += (acc / other) / sigg[(size_t)b * BSTEP + k];

        // advance h: h_new[j] = sum_i Pend[j][i] h[i], then renormalize
        const float* Pr = &pendg[(size_t)b * 1024 + lane * 32];
        float hv = 0.0f;
        #pragma unroll 4
        for (int i = 0; i < 32; ++i) hv += Pr[i] * hl[i];
        float s = hv;
        #pragma unroll
        for (int m = 16; m >= 1; m >>= 1) s += __shfl_xor(s, m);
        __syncthreads();
        hl[lane] = hv / s;
        __syncthreads();
    }
    #pragma unroll
    for (int m = 16; m >= 1; m >>= 1) lacc += __shfl_xor(lacc, m);
    if (lane == 0) out[0] = lacc;
}

// ---------------------------------------------------------------------------
extern "C" void kernel_launch(void* const* d_in, const int* in_sizes, int n_in,
                              void* d_out, int out_size, void* d_ws, size_t ws_size,
                              hipStream_t stream) {
    const float* obs2 = (const float*)d_in[0];
    const float* obs1 = (const float*)d_in[1];
    const float* mean = (const float*)d_in[2];
    const float* var  = (const float*)d_in[3];
    const float* bate = (const float*)d_in[4];
    const float* pi   = (const float*)d_in[5];
    const float* aij  = (const float*)d_in[6];

    const int T  = in_sizes[0];
    const int NB = (T + BSTEP - 1) / BSTEP;

    float* ws   = (float*)d_ws;
    float* Mtg  = ws;
    float* M2g  = Mtg  + 1024;
    float* qhat = M2g  + 1024;
    float* rhat = qhat + (size_t)NB * BSTEP * 32;
    float* sig  = rhat + (size_t)NB * BSTEP * 32;
    float* pend = sig  + (size_t)NB * BSTEP;

    ms_ar_precompute<<<1, 1024, 0, stream>>>(mean, var, bate, aij, Mtg, M2g);
    ms_ar_pass1<<<NB, 128, 0, stream>>>(obs2, obs1, var, bate, Mtg, M2g,
                                        qhat, rhat, sig, pend, T);
    ms_ar_pass2<<<1, 32, 0, stream>>>(pi, qhat, rhat, sig, pend,
                                      (float*)d_out, NB, T);
    (void)n_in; (void)out_size; (void)ws_size;
}